// RelationModel_83167746720086
// MI455X (gfx1250) — compile-verified
//
#include <hip/hip_runtime.h>
#include <math.h>

// ---------------- problem constants ----------------
constexpr int kB  = 8;      // batch
constexpr int kN  = 128;    // objects
constexpr int kA  = 4;      // attrs
constexpr int kT  = 12;     // program steps
constexpr int kMC = 2048;   // MAX_CONCEPTS
constexpr int kC  = kMC + kN;       // 2176 concept dim
constexpr int kE  = 128;    // EMBED_DIM
constexpr int kID = 64;     // IDENTITY_DIM
constexpr int kAD = 64;     // ATTENTION_DIM
constexpr int kH  = 256;    // HIDDEN_DIM

typedef __attribute__((ext_vector_type(2))) float v2f;
typedef __attribute__((ext_vector_type(8))) float v8f;

// fp32 WMMA: D(16x16) += A(16x4) * B(4x16), all f32 (exact fp32 path, RNE).
// A layout: lane (r = lane&15, half = lane>>4): a[i] = A[r, 2*half + i]
// B layout: b[i] = B[2*half + i, r]
// C/D layout: acc[v] = D[v + 8*half, r]
__device__ __forceinline__ v8f wmma_f32(v2f a, v2f b, v8f c) {
    return __builtin_amdgcn_wmma_f32_16x16x4_f32(false, a, false, b, (short)0, c,
                                                 false, false);
}

// ---------------- build dendron / axon / identity / attention0 ----------------
// grid = kB*kC blocks, 128 threads (thread = embed index e)
__global__ void build_scene_k(const int* __restrict__ scene,
                              const float* __restrict__ aein,
                              const float* __restrict__ aeout,
                              const float* __restrict__ aeid,
                              const float* __restrict__ cein,
                              const float* __restrict__ ceout,
                              const float* __restrict__ ceid,
                              const float* __restrict__ att_init,
                              float* __restrict__ dendron,
                              float* __restrict__ axon,
                              float* __restrict__ identity,
                              float* __restrict__ attention) {
    int idx = blockIdx.x;
    int b = idx / kC, c = idx % kC;
    int e = threadIdx.x;  // 0..127
    float din, dout, idv = 0.0f;
    if (c < kMC) {
        din  = cein[c * kE + e];
        dout = ceout[c * kE + e];
        if (e < kID) idv = ceid[c * kID + e];
    } else {
        int n = c - kMC;
        float si = 0.f, so = 0.f, sid = 0.f;
        for (int a = 0; a < kA; ++a) {
            int s = scene[(b * kN + n) * kA + a];
            float m = (s != -1) ? 1.0f : 0.0f;
            int row = s + 1;
            si += m * aein[row * kE + e];
            so += m * aeout[row * kE + e];
            if (e < kID) sid += m * aeid[row * kID + e];
        }
        din = si; dout = so; idv = sid;
    }
    size_t r = (size_t)idx;
    dendron[r * kE + e] = din;
    axon[r * kE + e]    = dout;
    if (e < kID) {
        identity[r * kID + e]  = idv;
        attention[r * kAD + e] = att_init[e];
    }
}

// meta[b,a] = attention_init[a]
__global__ void init_meta_k(const float* __restrict__ att_init, float* __restrict__ meta) {
    int t = threadIdx.x;  // 512 threads = kB*kAD
    if (t < kB * kAD) meta[t] = att_init[t % kAD];
}

// ---------------- tiled transpose: XT (kE x R) = X^T, X is (R x kE) row-major -----
// grid (R/32, kE/32, kB), block (32, 8)
__global__ void transpose_k(const float* __restrict__ X, float* __restrict__ XT, int R) {
    __shared__ float tile[32][33];
    int b = blockIdx.z;
    const float* Xb = X + (size_t)b * R * kE;
    float* XTb = XT + (size_t)b * kE * R;
    int r0 = blockIdx.x * 32;  // row block in X
    int c0 = blockIdx.y * 32;  // col block in X
    int tx = threadIdx.x;      // 0..31
    int ty = threadIdx.y;      // 0..7
    for (int i = ty; i < 32; i += 8)
        tile[i][tx] = Xb[(size_t)(r0 + i) * kE + c0 + tx];
    __syncthreads();
    for (int i = ty; i < 32; i += 8)
        XTb[(size_t)(c0 + i) * R + r0 + tx] = tile[tx][i];
}

// ---------------- meta MLP step (tiny, one block per batch) ----------------
__global__ void meta_step_k(const float* __restrict__ ceout,
                            const int* __restrict__ prog_op,
                            const int* __restrict__ prog_arg,
                            const float* __restrict__ mw1, const float* __restrict__ mb1,
                            const float* __restrict__ mw2, const float* __restrict__ mb2,
                            float* __restrict__ meta, int t) {
    int b = blockIdx.x;
    __shared__ float xin[kAD + kE];  // [meta(64), arg(128)]
    __shared__ float h[kH];
    __shared__ float nm[kAD];
    int tid = threadIdx.x;  // 256
    int arg = prog_arg[b * kT + t];
    if (tid < kAD) xin[tid] = meta[b * kAD + tid];
    else if (tid < kAD + kE) xin[tid] = ceout[arg * kE + (tid - kAD)];
    __syncthreads();
    float acc = mb1[tid];
    for (int k = 0; k < kAD + kE; ++k) acc += mw1[tid * (kAD + kE) + k] * xin[k];
    h[tid] = fmaxf(acc, 0.0f);
    __syncthreads();
    if (tid < kAD) {
        float o = mb2[tid];
        for (int k = 0; k < kH; ++k) o += mw2[tid * kH + k] * h[k];
        nm[tid] = o;
    }
    __syncthreads();
    if (tid < kAD) {
        float m = (prog_op[b * kT + t] == 0) ? 1.0f : 0.0f;
        meta[b * kAD + tid] = m * nm[tid] + (1.0f - m) * meta[b * kAD + tid];
    }
}

// ---------------- per-row scale: s, proj, ia_scaled ----------------
// grid = kB*kC blocks, 128 threads (d = 0..127)
__global__ void scale_step_k(const float* __restrict__ ceout,
                             const int* __restrict__ prog_arg,
                             const float* __restrict__ axon,
                             const float* __restrict__ identity,
                             const float* __restrict__ attention,
                             float* __restrict__ ia_scaled,
                             float* __restrict__ proj, int t) {
    int idx = blockIdx.x;
    int b = idx / kC;
    int d = threadIdx.x;
    __shared__ float red[128];
    const float* arow = attention + (size_t)idx * kAD;
    // s = mean_a attention[b,c,a]
    red[d] = (d < kAD) ? arow[d] : 0.0f;
    __syncthreads();
    for (int off = 64; off > 0; off >>= 1) {
        if (d < off) red[d] += red[d + off];
        __syncthreads();
    }
    float s = red[0] * (1.0f / (float)kAD);
    __syncthreads();
    // proj = (arg . axon_row) / E
    int arg = prog_arg[b * kT + t];
    float av = ceout[arg * kE + d];
    float xr = axon[(size_t)idx * kE + d];
    red[d] = av * xr;
    __syncthreads();
    for (int off = 64; off > 0; off >>= 1) {
        if (d < off) red[d] += red[d + off];
        __syncthreads();
    }
    if (d == 0) proj[idx] = red[0] * (1.0f / (float)kE);
    // ia_scaled[b,c,d] = concat(identity, attention)[d] * s
    float iav = (d < kID) ? identity[(size_t)idx * kID + d] : arow[d - kID];
    ia_scaled[(size_t)idx * (kID + kAD) + d] = iav * s;
}

// ---------------- M^T = (axon^T @ ia_scaled)^T with /C folded -----------------
// Inputs are pre-transposed so every fragment load is a contiguous v2f (b64):
//   axonT (kE x C): A[m,k] = axonT[m*C + k]
//   iaT   (kE x C): B[k,n] = iaT[n*C + k]
// Output stored transposed: Mt[n*kE + m] = M[m,n], so the next GEMM's B loads
// are contiguous as well. grid (8, 8, kB), 32 threads, K = C.
__global__ void gemm_M_k(const float* __restrict__ axonT,
                         const float* __restrict__ iaT,
                         float* __restrict__ Mt) {
    int b = blockIdx.z;
    int m0 = blockIdx.x * 16, n0 = blockIdx.y * 16;
    const float* At = axonT + (size_t)b * kE * kC;
    const float* Bt = iaT + (size_t)b * kE * kC;
    int lane = threadIdx.x;
    int r = lane & 15, kk = (lane >> 4) * 2;
    const float* ap = At + (size_t)(m0 + r) * kC + kk;
    const float* bp = Bt + (size_t)(n0 + r) * kC + kk;
    v8f acc = {};
    for (int k0 = 0; k0 < kC; k0 += 4) {
        v2f a  = *(const v2f*)(ap + k0);
        v2f bv = *(const v2f*)(bp + k0);
        acc = wmma_f32(a, bv, acc);
    }
    const float sc = 1.0f / (float)kC;
    float* Mo = Mt + (size_t)b * kE * kE;
    int half = lane >> 4;
    for (int v = 0; v < 8; ++v)
        Mo[(n0 + r) * kE + m0 + v + 8 * half] = acc[v] * sc;
}

// ---------------- gathered[b] = dendron[b] (C x 128) @ M[b] (128 x 128) ----------
// 16x128 strip per wave: 8 accumulators, A fragment reused 8x.
// grid (136, 1, kB), 32 threads, K = 128. B[k,n] = Mt[n*kE + k] (contiguous v2f).
__global__ void gemm_gather_k(const float* __restrict__ dendron,
                              const float* __restrict__ Mt,
                              float* __restrict__ gathered) {
    int b = blockIdx.z;
    int m0 = blockIdx.x * 16;
    const float* Ad = dendron + (size_t)b * kC * kE;  // row-major (C,128)
    const float* Bm = Mt + (size_t)b * kE * kE;
    int lane = threadIdx.x;
    int r = lane & 15, kk = (lane >> 4) * 2, half = lane >> 4;
    v8f acc[8];
#pragma unroll
    for (int nt = 0; nt < 8; ++nt) acc[nt] = {};
    const float* ap = Ad + (size_t)(m0 + r) * kE + kk;
    for (int k0 = 0; k0 < kE; k0 += 4) {
        v2f a = *(const v2f*)(ap + k0);
#pragma unroll
        for (int nt = 0; nt < 8; ++nt) {
            v2f bv = *(const v2f*)(Bm + (nt * 16 + r) * kE + k0 + kk);
            acc[nt] = wmma_f32(a, bv, acc[nt]);
        }
    }
    float* G = gathered + (size_t)b * kC * kE;
#pragma unroll
    for (int nt = 0; nt < 8; ++nt)
        for (int v = 0; v < 8; ++v)
            G[(size_t)(m0 + v + 8 * half) * kE + nt * 16 + r] = acc[nt][v];
}

// ---------------- hidden = relu(gathered @ w1^T + b1), (B*C,128)->(B*C,256) --------
// 16x128 strip per wave (8 accs). grid (136, 2, kB), 32 threads, K = 128.
// B[k,n] = w1[n,k] (w1 row-major (256,128)) -> contiguous v2f.
__global__ void mlp1_k(const float* __restrict__ gathered,
                       const float* __restrict__ aw1, const float* __restrict__ ab1,
                       float* __restrict__ hidden) {
    int b = blockIdx.z;
    int m0 = blockIdx.x * 16;
    int nbase = blockIdx.y * 128;
    const float* X = gathered + (size_t)b * kC * kE;
    int lane = threadIdx.x;
    int r = lane & 15, kk = (lane >> 4) * 2, half = lane >> 4;
    v8f acc[8];
#pragma unroll
    for (int nt = 0; nt < 8; ++nt) acc[nt] = {};
    const float* ap = X + (size_t)(m0 + r) * kE + kk;
    for (int k0 = 0; k0 < kE; k0 += 4) {
        v2f a = *(const v2f*)(ap + k0);
#pragma unroll
        for (int nt = 0; nt < 8; ++nt) {
            v2f bv = *(const v2f*)(aw1 + (size_t)(nbase + nt * 16 + r) * kE + k0 + kk);
            acc[nt] = wmma_f32(a, bv, acc[nt]);
        }
    }
    float* H = hidden + (size_t)b * kC * kH;
#pragma unroll
    for (int nt = 0; nt < 8; ++nt) {
        int n = nbase + nt * 16 + r;
        float bias = ab1[n];
        for (int v = 0; v < 8; ++v)
            H[(size_t)(m0 + v + 8 * half) * kH + n] = fmaxf(acc[nt][v] + bias, 0.0f);
    }
}

// ---------------- layer2 + attention update + history writes ----------------
// 16x64 strip per wave (4 accs). grid (136, 1, kB), 32 threads, K = 256.
// B[k,n] = w2[n,k] (w2 row-major (64,256)) -> contiguous v2f.
__global__ void mlp2_update_k(const float* __restrict__ hidden,
                              const float* __restrict__ aw2, const float* __restrict__ ab2,
                              const float* __restrict__ gathered,
                              const float* __restrict__ meta,
                              const float* __restrict__ proj,
                              const int* __restrict__ prog_op,
                              float* __restrict__ attention,
                              float* __restrict__ att_hist,
                              float* __restrict__ ins_hist,
                              float* __restrict__ trans_hist, int t) {
    int b = blockIdx.z;
    int m0 = blockIdx.x * 16;
    const float* H = hidden + (size_t)b * kC * kH;
    int lane = threadIdx.x;
    int r = lane & 15, kk = (lane >> 4) * 2, half = lane >> 4;
    v8f acc[4];
#pragma unroll
    for (int nt = 0; nt < 4; ++nt) acc[nt] = {};
    const float* ap = H + (size_t)(m0 + r) * kH + kk;
    for (int k0 = 0; k0 < kH; k0 += 4) {
        v2f a = *(const v2f*)(ap + k0);
#pragma unroll
        for (int nt = 0; nt < 4; ++nt) {
            v2f bv = *(const v2f*)(aw2 + (size_t)(nt * 16 + r) * kH + k0 + kk);
            acc[nt] = wmma_f32(a, bv, acc[nt]);
        }
    }
    int op = prog_op[b * kT + t];
    float insg = (op == 1) ? 1.0f : 0.0f;
    float trg  = (op == 2) ? 1.0f : 0.0f;
    const float* G = gathered + (size_t)b * kC * kE;
    size_t hbase = (size_t)t * kB * kC * kAD + (size_t)b * kC * kAD;
#pragma unroll
    for (int nt = 0; nt < 4; ++nt) {
        int n = nt * 16 + r;
        float bias = ab2[n];
        float mval = meta[b * kAD + n];
        for (int v = 0; v < 8; ++v) {
            int m = m0 + v + 8 * half;
            float transfer = acc[nt][v] + bias + G[(size_t)m * kE + kID + n];
            float insert = mval * proj[b * kC + m];
            size_t ai = (size_t)b * kC * kAD + (size_t)m * kAD + n;
            float att = attention[ai] + insg * insert + trg * transfer;
            att = (att >= 0.0f) ? att : 0.01f * att;  // leaky
            att = fminf(fmaxf(att, -1.0f), 2.0f);     // clip
            attention[ai] = att;
            att_hist[hbase + (size_t)m * kAD + n]   = att;
            ins_hist[hbase + (size_t)m * kAD + n]   = insert;
            trans_hist[hbase + (size_t)m * kAD + n] = transfer;
        }
    }
}

// ---------------- log_softmax(attention.mean(2), axis=1) ----------------
__global__ void logsoftmax_k(const float* __restrict__ attention, float* __restrict__ out) {
    int b = blockIdx.x;
    int tid = threadIdx.x;  // 256
    __shared__ float xv[kC];
    __shared__ float red[256];
    for (int c = tid; c < kC; c += 256) {
        const float* a = attention + ((size_t)b * kC + c) * kAD;
        float s = 0.0f;
        for (int k = 0; k < kAD; ++k) s += a[k];
        xv[c] = s * (1.0f / (float)kAD);
    }
    __syncthreads();
    float mx = -INFINITY;
    for (int c = tid; c < kC; c += 256) mx = fmaxf(mx, xv[c]);
    red[tid] = mx;
    __syncthreads();
    for (int off = 128; off > 0; off >>= 1) {
        if (tid < off) red[tid] = fmaxf(red[tid], red[tid + off]);
        __syncthreads();
    }
    mx = red[0];
    __syncthreads();
    float se = 0.0f;
    for (int c = tid; c < kC; c += 256) se += expf(xv[c] - mx);
    red[tid] = se;
    __syncthreads();
    for (int off = 128; off > 0; off >>= 1) {
        if (tid < off) red[tid] += red[tid + off];
        __syncthreads();
    }
    float lse = logf(red[0]) + mx;
    for (int c = tid; c < kC; c += 256) out[b * kC + c] = xv[c] - lse;
}

// ---------------- host ----------------
extern "C" void kernel_launch(void* const* d_in, const int* in_sizes, int n_in,
                              void* d_out, int out_size, void* d_ws, size_t ws_size,
                              hipStream_t stream) {
    const int*   scene    = (const int*)d_in[0];
    const int*   prog_op  = (const int*)d_in[1];
    const int*   prog_arg = (const int*)d_in[2];
    const float* aein     = (const float*)d_in[3];
    const float* aeout    = (const float*)d_in[4];
    const float* aeid     = (const float*)d_in[5];
    const float* cein     = (const float*)d_in[6];
    const float* ceout    = (const float*)d_in[7];
    const float* ceid     = (const float*)d_in[8];
    // d_in[9] embed_init, d_in[10] identity_init: unused by the reference
    const float* att_init = (const float*)d_in[11];
    const float* aw1      = (const float*)d_in[12];
    const float* ab1      = (const float*)d_in[13];
    const float* aw2      = (const float*)d_in[14];
    const float* ab2      = (const float*)d_in[15];
    const float* mw1      = (const float*)d_in[16];
    const float* mb1      = (const float*)d_in[17];
    const float* mw2      = (const float*)d_in[18];
    const float* mb2      = (const float*)d_in[19];

    // workspace carve-out (floats); total ~80 MB
    float* ws = (float*)d_ws;
    size_t off = 0;
    float* dendron   = ws + off; off += (size_t)kB * kC * kE;   // 2,228,224
    float* axon      = ws + off; off += (size_t)kB * kC * kE;
    float* axonT     = ws + off; off += (size_t)kB * kE * kC;
    float* identity  = ws + off; off += (size_t)kB * kC * kID;  // 1,114,112
    float* attention = ws + off; off += (size_t)kB * kC * kAD;
    float* meta      = ws + off; off += (size_t)kB * kAD;
    float* ia        = ws + off; off += (size_t)kB * kC * (kID + kAD);
    float* iaT       = ws + off; off += (size_t)kB * kE * kC;
    float* Mt        = ws + off; off += (size_t)kB * kE * kE;
    float* gathered  = ws + off; off += (size_t)kB * kC * kE;
    float* hidden    = ws + off; off += (size_t)kB * kC * kH;
    float* proj      = ws + off; off += (size_t)kB * kC;

    // outputs, concatenated flat in return order
    float* out0       = (float*)d_out;
    float* att_hist   = out0 + (size_t)kB * kC;
    float* ins_hist   = att_hist + (size_t)kT * kB * kC * kAD;
    float* trans_hist = ins_hist + (size_t)kT * kB * kC * kAD;

    build_scene_k<<<kB * kC, 128, 0, stream>>>(scene, aein, aeout, aeid, cein, ceout,
                                               ceid, att_init, dendron, axon, identity,
                                               attention);
    init_meta_k<<<1, 512, 0, stream>>>(att_init, meta);
    // axonT built once per call
    transpose_k<<<dim3(kC / 32, kE / 32, kB), dim3(32, 8), 0, stream>>>(axon, axonT, kC);

    for (int t = 0; t < kT; ++t) {
        meta_step_k<<<kB, kH, 0, stream>>>(ceout, prog_op, prog_arg, mw1, mb1, mw2, mb2,
                                           meta, t);
        scale_step_k<<<kB * kC, 128, 0, stream>>>(ceout, prog_arg, axon, identity,
                                                  attention, ia, proj, t);
        transpose_k<<<dim3(kC / 32, kE / 32, kB), dim3(32, 8), 0, stream>>>(ia, iaT, kC);
        gemm_M_k<<<dim3(8, 8, kB), 32, 0, stream>>>(axonT, iaT, Mt);
        gemm_gather_k<<<dim3(kC / 16, 1, kB), 32, 0, stream>>>(dendron, Mt, gathered);
        mlp1_k<<<dim3(kC / 16, 2, kB), 32, 0, stream>>>(gathered, aw1, ab1, hidden);
        mlp2_update_k<<<dim3(kC / 16, 1, kB), 32, 0, stream>>>(
            hidden, aw2, ab2, gathered, meta, proj, prog_op, attention, att_hist,
            ins_hist, trans_hist, t);
    }
    logsoftmax_k<<<kB, 256, 0, stream>>>(attention, out0);
}